// LSTMDecoder_28080496181536
// MI455X (gfx1250) — compile-verified
//
#include <hip/hip_runtime.h>
#include <math.h>

#define FEATURE_DIM 2048
#define EMBED_DIM   512
#define HIDDEN_DIM  512
#define VOCAB_SIZE  10000
#define ATT_DIM     512
#define BATCH       32
#define NUM_PIXELS  196
#define MAX_LEN     20

typedef __attribute__((ext_vector_type(2))) float v2f;
typedef __attribute__((ext_vector_type(8))) float v8f;

// ---------------------------------------------------------------------------
// V_WMMA_F32_16X16X4_F32 per-lane layout (wave32):
//   A 16x4:  lane<16 -> row=lane, K=k+{0,1};  lane>=16 -> row=lane-16, K=k+{2,3}
//   B 4x16:  lane<16 -> col=lane, K=k+{0,1};  lane>=16 -> col=lane-16, K=k+{2,3}
//   C 16x16: vgpr r -> row r (lanes 0-15) / 8+r (lanes 16-31), col = lane&15
// Requires K % 4 == 0, lda/ldb even (aligned float2 loads).
// ---------------------------------------------------------------------------

template <int MT, int NT>
__device__ __forceinline__ void wmma_k_loop(
    v8f (&acc)[MT][NT],
    const float* __restrict__ A, int lda,
    const float* __restrict__ B, int ldb,
    int m0, int n0, int row16, int khalf, int K)
{
    const float* Ap[MT];
#pragma unroll
    for (int mi = 0; mi < MT; ++mi)
        Ap[mi] = A + (size_t)(m0 + mi * 16 + row16) * lda + khalf;
    const float* Bp[NT];
#pragma unroll
    for (int j = 0; j < NT; ++j)
        Bp[j] = B + (size_t)(n0 + j * 16 + row16) * ldb + khalf;

#pragma unroll 2
    for (int k = 0; k < K; k += 4) {
        v2f av[MT];
        v2f bv[NT];
#pragma unroll
        for (int mi = 0; mi < MT; ++mi) av[mi] = *(const v2f*)(Ap[mi] + k);
#pragma unroll
        for (int j = 0; j < NT; ++j)    bv[j]  = *(const v2f*)(Bp[j] + k);
#pragma unroll
        for (int mi = 0; mi < MT; ++mi)
#pragma unroll
            for (int j = 0; j < NT; ++j)
                acc[mi][j] = __builtin_amdgcn_wmma_f32_16x16x4_f32(
                    false, av[mi], false, bv[j], (short)0, acc[mi][j],
                    false, false);
    }
}

// Generic TN GEMM:  C[m, n] = sum_k A[m*lda+k] * B[n*ldb+k]  + bias[n]
// One wave per block computes a (16*MT) x (16*NT) block. No bounds checks:
// launches must cover exact multiples (tails via offset pointers + smaller NT).
template <int MT, int NT>
__global__ __launch_bounds__(32) void wmma_gemm_tn(
    const float* __restrict__ A, int lda,
    const float* __restrict__ B, int ldb,
    float* __restrict__ C, long long ldc, long long c_off,
    const float* __restrict__ bias, int K)
{
    const int lane  = threadIdx.x;          // full wave, EXEC all ones
    const int m0    = blockIdx.y * (16 * MT);
    const int n0    = blockIdx.x * (16 * NT);
    const int row16 = lane & 15;
    const int khalf = (lane >> 4) << 1;     // 0 or 2

    v8f acc[MT][NT];
#pragma unroll
    for (int mi = 0; mi < MT; ++mi)
#pragma unroll
        for (int j = 0; j < NT; ++j) acc[mi][j] = (v8f)0.0f;

    wmma_k_loop<MT, NT>(acc, A, lda, B, ldb, m0, n0, row16, khalf, K);

    const int crowoff = (lane >> 4) << 3;   // 0 or 8
#pragma unroll
    for (int mi = 0; mi < MT; ++mi) {
#pragma unroll
        for (int j = 0; j < NT; ++j) {
            const int ccol = n0 + j * 16 + row16;
            const float bval = bias[ccol];
#pragma unroll
            for (int r = 0; r < 8; ++r) {
                const long long idx =
                    c_off + (long long)(m0 + mi * 16 + crowoff + r) * ldc + ccol;
                C[idx] = acc[mi][j][r] + bval;
            }
        }
    }
}

// Dual-segment TN GEMM (fused LSTM gates):
//   C[m, n] = sum_k1 A1[m,k1]*B1[n,k1] + sum_k2 A2[m,k2]*B2[n,k2] + bias1[n] + bias2[n]
// Both K segments accumulate into the same WMMA accumulators; single epilogue.
template <int MT, int NT>
__global__ __launch_bounds__(32) void wmma_gemm_tn_dual(
    const float* __restrict__ A1, int lda1,
    const float* __restrict__ B1, int ldb1, int K1,
    const float* __restrict__ A2, int lda2,
    const float* __restrict__ B2, int ldb2, int K2,
    float* __restrict__ C, long long ldc,
    const float* __restrict__ bias1, const float* __restrict__ bias2)
{
    const int lane  = threadIdx.x;
    const int m0    = blockIdx.y * (16 * MT);
    const int n0    = blockIdx.x * (16 * NT);
    const int row16 = lane & 15;
    const int khalf = (lane >> 4) << 1;

    v8f acc[MT][NT];
#pragma unroll
    for (int mi = 0; mi < MT; ++mi)
#pragma unroll
        for (int j = 0; j < NT; ++j) acc[mi][j] = (v8f)0.0f;

    wmma_k_loop<MT, NT>(acc, A1, lda1, B1, ldb1, m0, n0, row16, khalf, K1);
    wmma_k_loop<MT, NT>(acc, A2, lda2, B2, ldb2, m0, n0, row16, khalf, K2);

    const int crowoff = (lane >> 4) << 3;
#pragma unroll
    for (int mi = 0; mi < MT; ++mi) {
#pragma unroll
        for (int j = 0; j < NT; ++j) {
            const int ccol = n0 + j * 16 + row16;
            const float bval = bias1[ccol] + bias2[ccol];
#pragma unroll
            for (int r = 0; r < 8; ++r) {
                const long long idx =
                    (long long)(m0 + mi * 16 + crowoff + r) * ldc + ccol;
                C[idx] = acc[mi][j][r] + bval;
            }
        }
    }
}

// ---------------------------------------------------------------------------
// Attention scores: e[b,p] = full_b + sum_a relu(enc_proj[b,p,a]+dec_proj[b,a])*full_W[a]
// One wave per (b,p) row; 8 waves per block.
// ---------------------------------------------------------------------------
__global__ __launch_bounds__(256) void attn_score_kernel(
    const float* __restrict__ enc_proj, const float* __restrict__ dec_proj,
    const float* __restrict__ full_W, const float* __restrict__ full_b,
    float* __restrict__ e_out)
{
    const int row  = blockIdx.x * 8 + (threadIdx.x >> 5); // 0..6271
    const int lane = threadIdx.x & 31;
    const int b = row / NUM_PIXELS;
    const float* ep = enc_proj + (size_t)row * ATT_DIM;
    const float* dp = dec_proj + (size_t)b * ATT_DIM;
    float s = 0.0f;
    for (int a = lane; a < ATT_DIM; a += 32) {
        float v = ep[a] + dp[a];
        s += fmaxf(v, 0.0f) * full_W[a];
    }
#pragma unroll
    for (int off = 16; off > 0; off >>= 1) s += __shfl_xor(s, off, 32);
    if (lane == 0) e_out[row] = s + full_b[0];
}

// ---------------------------------------------------------------------------
// Softmax over P=196, one wave per batch row.
// ---------------------------------------------------------------------------
__global__ __launch_bounds__(32) void softmax196_kernel(
    const float* __restrict__ e_in, float* __restrict__ alpha)
{
    const int b = blockIdx.x;
    const int lane = threadIdx.x;
    float vals[7];
    float m = -1e30f;
#pragma unroll
    for (int i = 0; i < 7; ++i) {
        const int p = lane + i * 32;
        float v = (p < NUM_PIXELS) ? e_in[b * NUM_PIXELS + p] : -1e30f;
        vals[i] = v;
        m = fmaxf(m, v);
    }
#pragma unroll
    for (int off = 16; off > 0; off >>= 1) m = fmaxf(m, __shfl_xor(m, off, 32));
    float s = 0.0f;
#pragma unroll
    for (int i = 0; i < 7; ++i) {
        const int p = lane + i * 32;
        if (p < NUM_PIXELS) {
            float ex = expf(vals[i] - m);
            vals[i] = ex;
            s += ex;
        }
    }
#pragma unroll
    for (int off = 16; off > 0; off >>= 1) s += __shfl_xor(s, off, 32);
    const float inv = 1.0f / s;
#pragma unroll
    for (int i = 0; i < 7; ++i) {
        const int p = lane + i * 32;
        if (p < NUM_PIXELS) alpha[b * NUM_PIXELS + p] = vals[i] * inv;
    }
}

// ---------------------------------------------------------------------------
// Context: lstm_in[b, 512+f] = sum_p alpha[b,p] * features[b,p,f]
// One block per batch element; alpha staged in LDS; coalesced feature reads.
// ---------------------------------------------------------------------------
__global__ __launch_bounds__(256) void context_kernel(
    const float* __restrict__ alpha, const float* __restrict__ features,
    float* __restrict__ lstm_in)
{
    __shared__ float sa[NUM_PIXELS];
    const int b = blockIdx.x;
    for (int p = threadIdx.x; p < NUM_PIXELS; p += 256) sa[p] = alpha[b * NUM_PIXELS + p];
    __syncthreads();
    const float* fb = features + (size_t)b * NUM_PIXELS * FEATURE_DIM;
    for (int f = threadIdx.x; f < FEATURE_DIM; f += 256) {
        float s = 0.0f;
        for (int p = 0; p < NUM_PIXELS; ++p)
            s += sa[p] * fb[(size_t)p * FEATURE_DIM + f];
        lstm_in[(size_t)b * (EMBED_DIM + FEATURE_DIM) + EMBED_DIM + f] = s;
    }
}

// ---------------------------------------------------------------------------
// Embedding gather for step t: lstm_in[b, 0:512] = emb[captions[b, t]]
// ---------------------------------------------------------------------------
__global__ __launch_bounds__(256) void gather_x_kernel(
    const int* __restrict__ captions, const float* __restrict__ emb,
    float* __restrict__ lstm_in, int t)
{
    const int i = blockIdx.x * 256 + threadIdx.x;   // 0 .. 32*512-1
    const int b  = i >> 9;
    const int ei = i & 511;
    const int tok = captions[b * MAX_LEN + t];
    lstm_in[(size_t)b * (EMBED_DIM + FEATURE_DIM) + ei] = emb[(size_t)tok * EMBED_DIM + ei];
}

// ---------------------------------------------------------------------------
// LSTM pointwise: gates [32, 2048] -> update c, h
// ---------------------------------------------------------------------------
__device__ __forceinline__ float sigmoidf_(float x) { return 1.0f / (1.0f + expf(-x)); }

__global__ __launch_bounds__(256) void lstm_pointwise_kernel(
    const float* __restrict__ gates, float* __restrict__ c, float* __restrict__ h)
{
    const int i = blockIdx.x * 256 + threadIdx.x;   // 0 .. 32*512-1
    const int b = i >> 9;
    const int j = i & 511;
    const float* g = gates + (size_t)b * (4 * HIDDEN_DIM);
    const float ig = sigmoidf_(g[j]);
    const float fg = sigmoidf_(g[HIDDEN_DIM + j]);
    const float gg = tanhf(g[2 * HIDDEN_DIM + j]);
    const float og = sigmoidf_(g[3 * HIDDEN_DIM + j]);
    const float cn = fg * c[i] + ig * gg;
    c[i] = cn;
    h[i] = og * tanhf(cn);
}

__global__ __launch_bounds__(256) void zero_kernel(float* __restrict__ p, int n)
{
    const int i = blockIdx.x * 256 + threadIdx.x;
    if (i < n) p[i] = 0.0f;
}

// ---------------------------------------------------------------------------
extern "C" void kernel_launch(void* const* d_in, const int* in_sizes, int n_in,
                              void* d_out, int out_size, void* d_ws, size_t ws_size,
                              hipStream_t stream)
{
    const float* features = (const float*)d_in[0];   // [32,196,2048]
    const int*   captions = (const int*)  d_in[1];   // [32,20]
    const float* emb      = (const float*)d_in[2];   // [10000,512]
    const float* W_ih     = (const float*)d_in[3];   // [2048,2560]
    const float* b_ih     = (const float*)d_in[4];   // [2048]
    const float* W_hh     = (const float*)d_in[5];   // [2048,512]
    const float* b_hh     = (const float*)d_in[6];   // [2048]
    const float* fc_W     = (const float*)d_in[7];   // [10000,512]
    const float* fc_b     = (const float*)d_in[8];   // [10000]
    const float* enc_W    = (const float*)d_in[9];   // [512,2048]
    const float* enc_b    = (const float*)d_in[10];  // [512]
    const float* dec_W    = (const float*)d_in[11];  // [512,512]
    const float* dec_b    = (const float*)d_in[12];  // [512]
    const float* full_W   = (const float*)d_in[13];  // [1,512]
    const float* full_b   = (const float*)d_in[14];  // [1]
    float* out = (float*)d_out;                      // [32,20,10000]

    // Scratch layout (floats), ~13.7 MB total
    float* ws        = (float*)d_ws;
    float* enc_proj  = ws;                                              // 6272*512
    float* lstm_in   = enc_proj + (size_t)BATCH * NUM_PIXELS * ATT_DIM; // 32*2560
    float* hbuf      = lstm_in  + (size_t)BATCH * (EMBED_DIM + FEATURE_DIM);
    float* cbuf      = hbuf     + (size_t)BATCH * HIDDEN_DIM;
    float* dec_proj  = cbuf     + (size_t)BATCH * HIDDEN_DIM;           // 32*512
    float* ebuf      = dec_proj + (size_t)BATCH * ATT_DIM;              // 32*196
    float* alpha     = ebuf     + (size_t)BATCH * NUM_PIXELS;           // 32*196
    float* gates     = alpha    + (size_t)BATCH * NUM_PIXELS;           // 32*2048

    const int LSTM_IN_W = EMBED_DIM + FEATURE_DIM;   // 2560
    const int VOCAB_MAIN = (VOCAB_SIZE / 64) * 64;   // 9984
    const long long OUT_LDC = (long long)MAX_LEN * VOCAB_SIZE;

    // ---- Precompute -------------------------------------------------------
    // h0 = c0 = 0 (hbuf and cbuf are contiguous)
    zero_kernel<<<(2 * BATCH * HIDDEN_DIM + 255) / 256, 256, 0, stream>>>(
        hbuf, 2 * BATCH * HIDDEN_DIM);

    // enc_proj = features @ enc_W^T + enc_b   (M=6272, N=512, K=2048)
    wmma_gemm_tn<2, 4><<<dim3(ATT_DIM / 64, (BATCH * NUM_PIXELS) / 32), 32, 0, stream>>>(
        features, FEATURE_DIM, enc_W, FEATURE_DIM,
        enc_proj, ATT_DIM, 0, enc_b, FEATURE_DIM);

    // ---- Time steps -------------------------------------------------------
    for (int t = 0; t < MAX_LEN; ++t) {
        // dec_proj = h @ dec_W^T + dec_b   (M=32, N=512, K=512)
        wmma_gemm_tn<2, 4><<<dim3(ATT_DIM / 64, 1), 32, 0, stream>>>(
            hbuf, HIDDEN_DIM, dec_W, HIDDEN_DIM,
            dec_proj, ATT_DIM, 0, dec_b, HIDDEN_DIM);

        // attention scores
        attn_score_kernel<<<(BATCH * NUM_PIXELS) / 8, 256, 0, stream>>>(
            enc_proj, dec_proj, full_W, full_b, ebuf);

        // softmax over pixels
        softmax196_kernel<<<BATCH, 32, 0, stream>>>(ebuf, alpha);

        // context -> lstm_in[:, 512:2560]
        context_kernel<<<BATCH, 256, 0, stream>>>(alpha, features, lstm_in);

        // x_t -> lstm_in[:, 0:512]
        gather_x_kernel<<<(BATCH * EMBED_DIM) / 256, 256, 0, stream>>>(
            captions, emb, lstm_in, t);

        // gates = lstm_in @ W_ih^T + h @ W_hh^T + b_ih + b_hh
        // (M=32, N=2048, K = 2560 then 512, fused dual-segment accumulation)
        wmma_gemm_tn_dual<2, 4><<<dim3((4 * HIDDEN_DIM) / 64, 1), 32, 0, stream>>>(
            lstm_in, LSTM_IN_W, W_ih, LSTM_IN_W, LSTM_IN_W,
            hbuf, HIDDEN_DIM, W_hh, HIDDEN_DIM, HIDDEN_DIM,
            gates, 4 * HIDDEN_DIM, b_ih, b_hh);

        // c, h update
        lstm_pointwise_kernel<<<(BATCH * HIDDEN_DIM) / 256, 256, 0, stream>>>(
            gates, cbuf, hbuf);

        // logits: out[b, t, :] = h @ fc_W^T + fc_b   (M=32, N=10000, K=512)
        // main body: 156 strips of 64 columns
        wmma_gemm_tn<2, 4><<<dim3(VOCAB_MAIN / 64, 1), 32, 0, stream>>>(
            hbuf, HIDDEN_DIM, fc_W, HIDDEN_DIM,
            out, OUT_LDC, (long long)t * VOCAB_SIZE, fc_b, HIDDEN_DIM);
        // tail: columns 9984..9999 (one 16-wide tile) via offset pointers
        wmma_gemm_tn<2, 1><<<dim3(1, 1), 32, 0, stream>>>(
            hbuf, HIDDEN_DIM, fc_W + (size_t)VOCAB_MAIN * HIDDEN_DIM, HIDDEN_DIM,
            out, OUT_LDC, (long long)t * VOCAB_SIZE + VOCAB_MAIN,
            fc_b + VOCAB_MAIN, HIDDEN_DIM);
    }
}